// AuxSpatialGatherModule_24910810317446
// MI455X (gfx1250) — compile-verified
//
#include <hip/hip_runtime.h>

typedef __attribute__((ext_vector_type(2))) float v2f;
typedef __attribute__((ext_vector_type(8))) float v8f;
typedef __attribute__((ext_vector_type(4))) unsigned int v4u;
typedef __attribute__((ext_vector_type(4))) int v4i;
typedef __attribute__((ext_vector_type(8))) int v8i;

#define BATCH 16
#define CH    512
#define HW    16384
#define KCLS  19
#define CT    4              // channel tiles per batch (128 channels each)
#define ST    8              // pixel slices per batch (2048 px each)
#define PIX_TILE 32
#define RS    35             // LDS row stride (words); 35 invertible mod 64 -> conflict-free
#define TILE_ROWS 128        // channels per workgroup tile
#define TILE_WORDS (TILE_ROWS * RS)       // 4480 words = 17920 B per buffer
#define SLICE (HW / ST)                   // 2048 px
#define NITER (SLICE / PIX_TILE)          // 64

#if __has_builtin(__builtin_amdgcn_tensor_load_to_lds)
#define HAVE_TDM 1
#else
#define HAVE_TDM 0
#endif

// ---------------- Kernel 1: zero output + per-class valid-pixel counts ----------------
__global__ __launch_bounds__(256) void asg_count_zero(const int* __restrict__ gt,
                                                      float* __restrict__ out,
                                                      int* __restrict__ counts) {
    int b = blockIdx.x;
    int t = threadIdx.x;
    const int per_b = CH * KCLS;
    float* ob = out + (size_t)b * per_b;
    for (int i = t; i < per_b; i += 256) ob[i] = 0.0f;

    __shared__ int cnt[32];
    if (t < 32) cnt[t] = 0;
    __syncthreads();
    const int* g = gt + (size_t)b * HW;
    for (int i = t; i < HW; i += 256) {
        int v = g[i];
        if (v != 255) {
            int m = v > (KCLS - 1) ? (KCLS - 1) : v;  // reference clips then masks
            atomicAdd(&cnt[m], 1);
        }
    }
    __syncthreads();
    if (t < 32) counts[b * 32 + t] = cnt[t];
}

// ---------------- TDM: one 2D descriptor loads [128 rows x 32 px] with LDS padding ----
__device__ __forceinline__ void tdm_issue(const float* src, unsigned lds_byte_off) {
#if HAVE_TDM
    unsigned long long ga = (unsigned long long)(uintptr_t)src;
    v4u g0;
    g0[0] = 1u;                                   // count=1, user load descriptor
    g0[1] = lds_byte_off;                         // lds_addr (bytes)
    g0[2] = (unsigned)ga;                         // global_addr[31:0]
    g0[3] = (unsigned)((ga >> 32) & 0x01FFFFFFu)  // global_addr[56:32]
          | (2u << 30);                           // type = 2 ("image")
    v8i g1;
    g1[0] = (int)((2u << 16)     // data_size = 4 bytes
                | (1u << 20)     // pad_enable: hardware bank-padding on LDS writes
                | (4u << 22)     // pad_interval = 32 DWORDs (one 32-px row)
                | (2u << 25));   // pad_amount  = 3 DWORDs -> row stride 35 words
    g1[1] = (int)(PIX_TILE << 16);    // tensor_dim0 = 32 (bits 63:48); barrier addr 0
    g1[2] = (int)(TILE_ROWS << 16);   // tensor_dim1 = 128 (bits 95:80)
    g1[3] = (int)(PIX_TILE << 16);    // tile_dim0 = 32 (bits 127:112)
    g1[4] = (int)(TILE_ROWS);         // tile_dim1 = 128; tile_dim2 = 0
    g1[5] = (int)(HW);                // tensor_dim0_stride[31:0] = 16384 elements
    g1[6] = 0;                        // stride MSBs / dim1 stride (unused, 2D)
    g1[7] = 0;
    v4i gz = {0, 0, 0, 0};
#if __clang_major__ >= 23
    v8i gz8 = {0, 0, 0, 0, 0, 0, 0, 0};
    __builtin_amdgcn_tensor_load_to_lds(g0, g1, gz, gz, gz8, 0);
#else
    __builtin_amdgcn_tensor_load_to_lds(g0, g1, gz, gz, 0);
#endif
#else
    (void)src; (void)lds_byte_off;
#endif
}

// ---------------- Kernel 2: one-hot x feats GEMM via V_WMMA_F32_16X16X4_F32 ------------
__global__ __launch_bounds__(256) void asg_wmma(const float* __restrict__ feats,
                                                const int* __restrict__ gt,
                                                float* __restrict__ out) {
    __shared__ float sm[2][TILE_WORDS];          // double-buffered DMA landing zone

    const int wid  = threadIdx.x >> 5;
    const int lane = threadIdx.x & 31;
    const int n16  = lane & 15;
    const int half = lane >> 4;

    const int b  = blockIdx.z;
    const int ct = blockIdx.y;
    const int st = blockIdx.x;

    const int cb = ct * TILE_ROWS;
    const float* fbase = feats + ((size_t)b * CH + cb) * HW + st * SLICE;
    const int*   gbase = gt + (size_t)b * HW + st * SLICE;

    // visible LDS stores (overwritten by DMA) so loads can't be folded away
    sm[0][threadIdx.x] = 0.0f;
    sm[1][threadIdx.x] = 0.0f;
    __syncthreads();

    v8f acc0 = {};   // classes 0..15  x 16 channels
    v8f acc1 = {};   // classes 16..31 (19..31 padding, dropped at writeout)

#if HAVE_TDM
    if (wid == 0) tdm_issue(fbase, 0u);          // prologue: fill buffer 0
#endif

    #pragma unroll 1
    for (int it = 0; it < NITER; ++it) {
        const int p0 = it * PIX_TILE;
        const int buf = it & 1;

#if HAVE_TDM
        if (wid == 0) {
            if (it + 1 < NITER) {
                tdm_issue(fbase + p0 + PIX_TILE, (unsigned)((buf ^ 1) * TILE_WORDS * 4));
                __builtin_amdgcn_s_wait_tensorcnt(1);   // current buffer complete
            } else {
                __builtin_amdgcn_s_wait_tensorcnt(0);
            }
        }
        __syncthreads();                                 // DMA data visible to all waves
        const float* srow = &sm[buf][wid * 16 * RS];     // this wave's 16 channel rows
#else
        // Fallback staging: coalesced loads + LDS transpose (per-wave private region)
        float* swst = &sm[0][wid * 16 * RS];
        #pragma unroll
        for (int r = 0; r < 16; ++r)
            swst[r * RS + lane] = fbase[(size_t)(wid * 16 + r) * HW + p0 + lane];
        const float* srow = swst;
#endif

        int gv = gbase[p0 + lane];                       // one gt pixel per lane

        #pragma unroll
        for (int j = 0; j < 8; ++j) {
            const int col = 4 * j + 2 * half;            // this lane-half's K pair
            int g0 = __builtin_amdgcn_ds_bpermute(col << 2, gv);
            int g1 = __builtin_amdgcn_ds_bpermute((col + 1) << 2, gv);
            int m0 = (g0 == 255) ? -1 : (g0 > (KCLS - 1) ? (KCLS - 1) : g0);
            int m1 = (g1 == 255) ? -1 : (g1 > (KCLS - 1) ? (KCLS - 1) : g1);

            v2f a0, a1, bm;
            a0.x = (m0 == n16)      ? 1.0f : 0.0f;
            a0.y = (m1 == n16)      ? 1.0f : 0.0f;
            a1.x = (m0 == n16 + 16) ? 1.0f : 0.0f;
            a1.y = (m1 == n16 + 16) ? 1.0f : 0.0f;
            bm.x = srow[n16 * RS + col];
            bm.y = srow[n16 * RS + col + 1];

            acc0 = __builtin_amdgcn_wmma_f32_16x16x4_f32(false, a0, false, bm,
                                                         (short)0, acc0, false, false);
            acc1 = __builtin_amdgcn_wmma_f32_16x16x4_f32(false, a1, false, bm,
                                                         (short)0, acc1, false, false);
        }

#if HAVE_TDM
        __syncthreads();   // all reads of this buffer done before next DMA overwrites it
#endif
    }

    // D layout: VGPR r, lanes 0-15 -> M=r, lanes 16-31 -> M=r+8; N = n16.
    const int c = cb + wid * 16 + n16;
    float* ob = out + ((size_t)b * CH + c) * KCLS;
    #pragma unroll
    for (int r = 0; r < 8; ++r) {
        int k0 = r + 8 * half;                 // 0..15 always valid
        atomicAdd(&ob[k0], acc0[r]);
        int k1 = 16 + k0;                      // only 16..18 valid
        if (k1 < KCLS) atomicAdd(&ob[k1], acc1[r]);
    }
}

// ---------------- Kernel 3: in-place normalization by per-class counts ----------------
__global__ __launch_bounds__(256) void asg_norm(float* __restrict__ out,
                                                const int* __restrict__ counts) {
    int idx = blockIdx.x * 256 + threadIdx.x;
    const int total = BATCH * CH * KCLS;
    if (idx >= total) return;
    int k = idx % KCLS;
    int b = idx / (CH * KCLS);
    int c = counts[b * 32 + k];
    float s = (c > 0) ? (1.0f / (float)c) : 1.0f;   // where(sum>0, sum, 1)
    out[idx] *= s;
}

extern "C" void kernel_launch(void* const* d_in, const int* in_sizes, int n_in,
                              void* d_out, int out_size, void* d_ws, size_t ws_size,
                              hipStream_t stream) {
    const float* feats = (const float*)d_in[0];
    const int*   gt    = (const int*)d_in[1];
    float* out = (float*)d_out;
    int* counts = (int*)d_ws;   // 16*32 ints

    asg_count_zero<<<dim3(BATCH), dim3(256), 0, stream>>>(gt, out, counts);
    asg_wmma<<<dim3(ST, CT, BATCH), dim3(256), 0, stream>>>(feats, gt, out);
    const int total = BATCH * CH * KCLS;
    asg_norm<<<dim3((total + 255) / 256), dim3(256), 0, stream>>>(out, counts);
}